// DynamicPruningMiniJMamba_15418932592969
// MI455X (gfx1250) — compile-verified
//
#include <hip/hip_runtime.h>
#include <math.h>

// Problem constants (B,S,D from reference)
#define B_DIM 8
#define S_DIM 4096
#define D_DIM 1024
#define CHUNKS 64                       // chunks per batch in pass 1
#define ROWS_PER_CHUNK (S_DIM / CHUNKS) // 64 rows per chunk
#define NTHREADS 256
#define NWAVES (NTHREADS / 32)

typedef float v2f __attribute__((ext_vector_type(2)));
typedef float v8f __attribute__((ext_vector_type(8)));

// ---------------------------------------------------------------------------
// Pass 1: per-(batch, chunk) partial channel sums and sums-of-squares.
// 512 blocks x 256 threads; each thread owns 4 channels (one float4 column).
// Deterministic: fixed sequential accumulation order, no atomics.
// ---------------------------------------------------------------------------
__global__ void __launch_bounds__(NTHREADS)
partial_stats_kernel(const float* __restrict__ x,
                     float* __restrict__ psum,   // [B*CHUNKS][D]
                     float* __restrict__ psq) {  // [B*CHUNKS][D]
  const int blk = blockIdx.x;          // 0..B*CHUNKS-1
  const int b   = blk >> 6;            // /CHUNKS
  const int c   = blk & (CHUNKS - 1);
  const int tid = threadIdx.x;
  const size_t base =
      ((size_t)b * S_DIM + (size_t)c * ROWS_PER_CHUNK) * D_DIM + (size_t)tid * 4;
  const float4* __restrict__ xp = (const float4*)(x + base);
  float4 s = {0.f, 0.f, 0.f, 0.f};
  float4 q = {0.f, 0.f, 0.f, 0.f};
#pragma unroll 4
  for (int r = 0; r < ROWS_PER_CHUNK; ++r) {
    float4 v = xp[(size_t)r * (D_DIM / 4)];
    s.x += v.x; s.y += v.y; s.z += v.z; s.w += v.w;
    q.x += v.x * v.x; q.y += v.y * v.y; q.z += v.z * v.z; q.w += v.w * v.w;
  }
  const size_t o = (size_t)blk * D_DIM + (size_t)tid * 4;
  *(float4*)(psum + o) = s;
  *(float4*)(psq  + o) = q;
}

// ---------------------------------------------------------------------------
// Pass 2 (single block): reduce partials -> global_stats + channel norms,
// per-batch norm means, soft sigmoid mask; then all 8 waves cooperate on the
// k-predictor MLP with f32 WMMA and write scalar k to the tail of d_out.
// ---------------------------------------------------------------------------
__global__ void __launch_bounds__(NTHREADS)
reduce_mask_k_kernel(const float* __restrict__ psum,
                     const float* __restrict__ psq,
                     const float* __restrict__ kp_w1,  // [1024][64]
                     const float* __restrict__ kp_b1,  // [64]
                     const float* __restrict__ kp_w2,  // [64]
                     const float* __restrict__ kp_b2,  // [1]
                     float* __restrict__ mask_out,     // [B][D]
                     float* __restrict__ out) {        // full output buffer
  __shared__ float s_gs[D_DIM];               // global channel means
  __shared__ float s_norm[B_DIM * D_DIM];     // per-batch channel L2 norms
  __shared__ float s_red[NTHREADS];
  __shared__ float s_part[NWAVES * 64];       // per-wave partial h
  __shared__ float s_dot[64];                 // relu(h)*w2 terms
  const int tid = threadIdx.x;

  // global_stats[d] = mean over (B,S)
  for (int d = tid; d < D_DIM; d += NTHREADS) {
    float acc = 0.f;
    for (int bc = 0; bc < B_DIM * CHUNKS; ++bc)
      acc += psum[(size_t)bc * D_DIM + d];
    s_gs[d] = acc * (1.f / (float)(B_DIM * S_DIM));
  }
  // channel_norms[b][d] = sqrt(sum_s x^2)
  for (int b = 0; b < B_DIM; ++b) {
    for (int d = tid; d < D_DIM; d += NTHREADS) {
      float acc = 0.f;
      for (int c = 0; c < CHUNKS; ++c)
        acc += psq[((size_t)(b * CHUNKS + c)) * D_DIM + d];
      s_norm[b * D_DIM + d] = sqrtf(acc);
    }
  }
  __syncthreads();

  // per-batch mean of norms, then soft mask = sigmoid(10*(norm - mean))
  for (int b = 0; b < B_DIM; ++b) {
    float local = 0.f;
    for (int d = tid; d < D_DIM; d += NTHREADS) local += s_norm[b * D_DIM + d];
    s_red[tid] = local;
    __syncthreads();
    for (int off = NTHREADS / 2; off > 0; off >>= 1) {
      if (tid < off) s_red[tid] += s_red[tid + off];
      __syncthreads();
    }
    const float mean = s_red[0] * (1.f / (float)D_DIM);
    __syncthreads();  // s_red reused next iteration
    for (int d = tid; d < D_DIM; d += NTHREADS) {
      float z = (s_norm[b * D_DIM + d] - mean) * 10.f;
      mask_out[b * D_DIM + d] = 1.f / (1.f + __expf(-z));
    }
  }

  // --- k predictor layer 1: h(64) = gs(1x1024) @ kp_w1(1024x64) -----------
  // All 8 waves run WMMA (EXEC all-ones per wave, no divergence). Each wave
  // owns a 128-wide K slice; 4 independent C accumulators (one per 16-wide
  // N tile) keep the 8 loads per step batched and the WMMAs independent.
  {
    const int wave = tid >> 5;
    const int lane = tid & 31;
    const int half = lane >> 4;                 // lanes 16-31 carry K+2/K+3
    const int lrow = lane & 15;                 // M for A, N-within-tile for B
    const float sel = (lrow == 0) ? 1.f : 0.f;  // only M row 0 is live
    v8f c0 = {}, c1 = {}, c2 = {}, c3 = {};
    const int kbeg = wave * (D_DIM / NWAVES);   // 128-wide slice
    for (int k0 = kbeg; k0 < kbeg + (D_DIM / NWAVES); k0 += 4) {
      const int ka = k0 + half * 2;             // VGPR0: K=0/2, VGPR1: K=1/3
      v2f a;
      a.x = s_gs[ka] * sel;
      a.y = s_gs[ka + 1] * sel;
      const float* w0 = kp_w1 + (size_t)ka * 64 + lrow;        // row K=ka
      const float* w1 = kp_w1 + (size_t)(ka + 1) * 64 + lrow;  // row K=ka+1
      v2f b0, b1, b2, b3;
      b0.x = w0[0];  b0.y = w1[0];
      b1.x = w0[16]; b1.y = w1[16];
      b2.x = w0[32]; b2.y = w1[32];
      b3.x = w0[48]; b3.y = w1[48];
      c0 = __builtin_amdgcn_wmma_f32_16x16x4_f32(false, a, false, b0, (short)0, c0, false, false);
      c1 = __builtin_amdgcn_wmma_f32_16x16x4_f32(false, a, false, b1, (short)0, c1, false, false);
      c2 = __builtin_amdgcn_wmma_f32_16x16x4_f32(false, a, false, b2, (short)0, c2, false, false);
      c3 = __builtin_amdgcn_wmma_f32_16x16x4_f32(false, a, false, b3, (short)0, c3, false, false);
    }
    // D layout: VGPR0, lanes 0-15 => D[M=0][N=lane]
    if (half == 0) {
      s_part[wave * 64 +  0 + lrow] = c0[0];
      s_part[wave * 64 + 16 + lrow] = c1[0];
      s_part[wave * 64 + 32 + lrow] = c2[0];
      s_part[wave * 64 + 48 + lrow] = c3[0];
    }
  }
  __syncthreads();

  // --- layer 2: k = clamp(sigmoid(relu(h+b1) . w2 + b2), 0.3, 1) ----------
  if (tid < 64) {
    float h = kp_b1[tid];
    for (int w = 0; w < NWAVES; ++w) h += s_part[w * 64 + tid];
    h = fmaxf(h, 0.f);
    s_dot[tid] = h * kp_w2[tid];
  }
  __syncthreads();
  if (tid < 32) {
    float acc = s_dot[tid] + s_dot[tid + 32];
    for (int off = 16; off > 0; off >>= 1) acc += __shfl_xor(acc, off, 32);
    if (tid == 0) {
      float kv = 1.f / (1.f + __expf(-(acc + kp_b2[0])));
      kv = fminf(fmaxf(kv, 0.3f), 1.f);
      out[(size_t)B_DIM * S_DIM * D_DIM] = kv;  // scalar k after pruned_x
    }
  }
}

// ---------------------------------------------------------------------------
// Pass 3: out[b,s,d] = x[b,s,d] * mask[b,d]  (pure streaming, float4)
// ---------------------------------------------------------------------------
__global__ void __launch_bounds__(NTHREADS)
apply_mask_kernel(const float* __restrict__ x,
                  const float* __restrict__ mask,  // [B][D]
                  float* __restrict__ out) {
  const size_t i = (size_t)blockIdx.x * NTHREADS + threadIdx.x;  // float4 idx
  const size_t e = i * 4;
  const int d = (int)(e & (D_DIM - 1));
  const int b = (int)(e >> 22);  // S_DIM*D_DIM = 2^22 elements per batch
  float4 xv = ((const float4*)x)[i];
  float4 mv = *(const float4*)(mask + ((size_t)b << 10) + d);
  float4 o;
  o.x = xv.x * mv.x; o.y = xv.y * mv.y; o.z = xv.z * mv.z; o.w = xv.w * mv.w;
  ((float4*)out)[i] = o;
}

// ---------------------------------------------------------------------------
extern "C" void kernel_launch(void* const* d_in, const int* in_sizes, int n_in,
                              void* d_out, int out_size, void* d_ws, size_t ws_size,
                              hipStream_t stream) {
  // setup_inputs order: x, sp_w1, sp_b1, sp_w2, sp_b2, kp_w1, kp_b1, kp_w2, kp_b2
  const float* x     = (const float*)d_in[0];
  const float* kp_w1 = (const float*)d_in[5];
  const float* kp_b1 = (const float*)d_in[6];
  const float* kp_w2 = (const float*)d_in[7];
  const float* kp_b2 = (const float*)d_in[8];
  float* out = (float*)d_out;

  // workspace layout (floats): partial sums | partial sumsq | mask
  float* ws    = (float*)d_ws;
  float* psum  = ws;                                    // B*CHUNKS*D = 512K f
  float* psq   = ws + (size_t)B_DIM * CHUNKS * D_DIM;   // 512K f
  float* maskb = psq + (size_t)B_DIM * CHUNKS * D_DIM;  // B*D = 8K f

  partial_stats_kernel<<<B_DIM * CHUNKS, NTHREADS, 0, stream>>>(x, psum, psq);
  reduce_mask_k_kernel<<<1, NTHREADS, 0, stream>>>(psum, psq, kp_w1, kp_b1,
                                                   kp_w2, kp_b2, maskb, out);
  const int n4 = (B_DIM * S_DIM * D_DIM) / 4;  // 8,388,608 float4s
  apply_mask_kernel<<<n4 / NTHREADS, NTHREADS, 0, stream>>>(x, maskb, out);
}